// calculate_geoLoss_4947802325291
// MI455X (gfx1250) — compile-verified
//
#include <hip/hip_runtime.h>
#include <hip/hip_bf16.h>

typedef __attribute__((ext_vector_type(2))) float v2f;
typedef __attribute__((ext_vector_type(8))) float v8f;

#define N_VERTS 35709
#define NLM     68
#define NROWS   204      // 68 landmarks * 3 coords
#define KA      80       // id_base K
#define KD      64       // exp_base K
#define KTOT    144      // 80 + 64
#define NBATCH  64
#define LDSW    145      // padded LDS row stride (odd -> bank-conflict-free)

__constant__ int c_lm_idx[NLM] = {
    27440, 27208, 27608, 27816, 35472, 34766, 34312, 34022, 33838, 33654,
    33375, 32939, 32244, 16264, 16467, 16888, 16644, 31716, 31056, 30662,
    30454, 30288, 29549, 29382, 29177, 28787, 28111,  8161,  8177,  8187,
     8192,  9883,  9163,  8204,  7243,  6515, 14066, 12383, 11353, 10455,
    11492, 12653,  5828,  4920,  3886,  2215,  3640,  4801, 10795, 10395,
     8935,  8215,  7495,  6025,  5522,  6915,  7636,  8236,  8836,  9555,
    10537,  9064,  8223,  7384,  5909,  7629,  8229,  8829
};

__global__ void geo_loss_zero_kernel(float* out) {
    if (threadIdx.x == 0) out[0] = 0.0f;
}

// grid = 13 blocks (M tiles of 16 rows), block = 128 threads = 4 wave32s
// wave w handles N-tile w (batches 16w .. 16w+15). K = 144 in steps of 4
// via V_WMMA_F32_16X16X4_F32 (full fp32 precision).
__global__ __launch_bounds__(128) void geo_loss_kernel(
    const float* __restrict__ alpha,
    const float* __restrict__ delta,
    const float* __restrict__ ref_alpha,
    const float* __restrict__ delta_pred,
    const float* __restrict__ id_base,   // [3*N_VERTS, 80] row-major
    const float* __restrict__ exp_base,  // [3*N_VERTS, 64] row-major
    float* __restrict__ out)
{
    __shared__ float sX[NBATCH * LDSW]; // X[b][k] = coefficient differences
    __shared__ float sW[16 * LDSW];     // gathered landmark rows for this M tile
    __shared__ float sAcc;

    const int tid   = threadIdx.x;
    const int mtile = blockIdx.x;

    if (tid == 0) sAcc = 0.0f;

    // ---- Stage X = concat(alpha - reference_alpha, delta - delta_pred) ----
    // 64*144 = 9216 elements / 128 threads = 72 each (uniform trip count).
    for (int idx = tid; idx < NBATCH * KTOT; idx += 128) {
        int b = idx / KTOT;
        int k = idx - b * KTOT;
        float v;
        if (k < KA) {
            v = alpha[b * KA + k] - ref_alpha[b * KA + k];
        } else {
            int kd = k - KA;
            v = delta[b * KD + kd] - delta_pred[b * KD + kd];
        }
        sX[b * LDSW + k] = v;
    }

    // ---- Stage W tile: 16 gathered rows of [id_base | exp_base] ----------
    // 16*144 = 2304 elements / 128 threads = 18 each (uniform trip count).
    for (int idx = tid; idx < 16 * KTOT; idx += 128) {
        int mr = idx / KTOT;
        int k  = idx - mr * KTOT;
        int r  = mtile * 16 + mr;            // global output row
        float v = 0.0f;
        if (r < NROWS) {
            int l   = r / 3;                 // landmark id
            int c   = r - l * 3;             // coord 0..2
            int row = 3 * c_lm_idx[l] + c;   // row into the 3*N_VERTS bases
            v = (k < KA) ? id_base[row * KA + k]
                         : exp_base[row * KD + (k - KA)];
        }
        sW[mr * LDSW + k] = v;
    }
    __syncthreads();

    // ---- WMMA: D[16 rows x 16 batches] = W_tile(16x144) * X^T(144x16) ----
    const int lane = tid & 31;
    const int wid  = tid >> 5;       // N tile index 0..3
    const int half = lane >> 4;      // 0: lanes 0-15, 1: lanes 16-31
    const int mn   = lane & 15;      // M for A-frag / N for B-frag
    const int n0   = wid * 16;

    const float* wrow = &sW[mn * LDSW];
    const float* xrow = &sX[(n0 + mn) * LDSW];

    v8f acc = {};
#pragma unroll
    for (int k0 = 0; k0 < KTOT; k0 += 4) {
        // fp32 A 16x4 layout (ISA 7.12.2): lane=M; VGPR0 K = 2*half, VGPR1 K = 2*half+1
        v2f a, b;
        a.x = wrow[k0 + 2 * half + 0];
        a.y = wrow[k0 + 2 * half + 1];
        // B 4x16 mirrors A: lane=N
        b.x = xrow[k0 + 2 * half + 0];
        b.y = xrow[k0 + 2 * half + 1];
        acc = __builtin_amdgcn_wmma_f32_16x16x4_f32(
                  /*neg_a=*/false, a, /*neg_b=*/false, b,
                  /*c_mod=*/(short)0, acc,
                  /*reuse_a=*/false, /*reuse_b=*/false);
    }

    // ---- Weighted square-sum epilogue -----------------------------------
    // C/D layout: lane%16 = N (batch), VGPR v -> M = v + 8*half.
    float local = 0.0f;
#pragma unroll
    for (int v = 0; v < 8; ++v) {
        int r = mtile * 16 + v + 8 * half;
        if (r < NROWS) {
            int   l = r / 3;
            float w = ((l >= 17 && l < 27) || (l >= 36)) ? 50.0f : 1.0f;
            float d = acc[v];
            local += w * d * d;
        }
    }
    atomicAdd(&sAcc, local);          // ds_add_f32
    __syncthreads();
    if (tid == 0) atomicAdd(out, sAcc * (1.0f / 68.0f));  // global_atomic_add_f32
}

extern "C" void kernel_launch(void* const* d_in, const int* in_sizes, int n_in,
                              void* d_out, int out_size, void* d_ws, size_t ws_size,
                              hipStream_t stream) {
    const float* alpha      = (const float*)d_in[0];
    const float* delta      = (const float*)d_in[1];
    const float* ref_alpha  = (const float*)d_in[2];
    const float* delta_pred = (const float*)d_in[3];
    const float* id_base    = (const float*)d_in[4];
    const float* exp_base   = (const float*)d_in[5];
    float* out = (float*)d_out;

    geo_loss_zero_kernel<<<1, 32, 0, stream>>>(out);

    // 204 output rows -> 13 M-tiles of 16; 128 threads = 4 wave32s = 4 N-tiles.
    geo_loss_kernel<<<13, 128, 0, stream>>>(alpha, delta, ref_alpha, delta_pred,
                                            id_base, exp_base, out);
}